// GatherLayer_1468878815558
// MI455X (gfx1250) — compile-verified
//
#include <hip/hip_runtime.h>

// GatherLayer: out[b, :] = full_output[b, idx[b]*512 : (idx[b]+1)*512]
// B=16384, OUTPUT_DIM=512, NB_ACTIONS=18.
// Pure streaming gather: ~64 MiB total HBM traffic -> ~2.9 us at 23.3 TB/s.
// Memory-bound; vector-memory (b128, non-temporal) path is the optimum on
// gfx1250. No matrix math exists in this op, so no WMMA is emitted by design.

#define GL_BATCH       16384
#define GL_OUTPUT_DIM  512
#define GL_NB_ACTIONS  18
#define GL_ROW_STRIDE  (GL_OUTPUT_DIM * GL_NB_ACTIONS)   // 9216 floats per input row
#define GL_VEC_PER_ROW (GL_OUTPUT_DIM / 4)               // 128 float4 per output row

// Native clang vector type: required by __builtin_nontemporal_* (HIP's float4
// is a class and is rejected). 16-byte, same layout as float4.
typedef float v4f __attribute__((ext_vector_type(4)));

__global__ __launch_bounds__(GL_VEC_PER_ROW) void
gather_rows_b128(const float* __restrict__ full,
                 const int*   __restrict__ indices,
                 float*       __restrict__ out)
{
    // One block per row: blockIdx.x is the row, so the index address is
    // wave-uniform -> scalar load. Each of the 128 threads moves one 16-byte
    // vector (GLOBAL_LOAD_B128 / GLOBAL_STORE_B128): whole 2 KB row per block.
    const unsigned row   = blockIdx.x;        // 0 .. 16383
    const unsigned chunk = threadIdx.x;       // 0 .. 127 (16B chunk in row)

    const int a = indices[row];               // 0 .. 17, uniform across block

    const v4f* __restrict__ src = reinterpret_cast<const v4f*>(
        full + (size_t)row * GL_ROW_STRIDE + (size_t)a * GL_OUTPUT_DIM);
    v4f* __restrict__ dst = reinterpret_cast<v4f*>(
        out + (size_t)row * GL_OUTPUT_DIM);

    // One-touch stream: non-temporal (TH=NT) on both sides so the 64 MiB
    // stream doesn't displace anything useful in WGP$/L2.
    v4f v = __builtin_nontemporal_load(src + chunk);
    __builtin_nontemporal_store(v, dst + chunk);
}

extern "C" void kernel_launch(void* const* d_in, const int* in_sizes, int n_in,
                              void* d_out, int out_size, void* d_ws, size_t ws_size,
                              hipStream_t stream)
{
    (void)in_sizes; (void)n_in; (void)d_ws; (void)ws_size; (void)out_size;

    const float* full    = (const float*)d_in[0];  // [16384, 9216] fp32
    const int*   indices = (const int*)d_in[1];    // [16384, 1] int32
    float*       out     = (float*)d_out;          // [16384, 512] fp32

    dim3 grid(GL_BATCH);            // one block per batch row
    dim3 block(GL_VEC_PER_ROW);     // 128 threads = 4 wave32s, one float4 each

    gather_rows_b128<<<grid, block, 0, stream>>>(full, indices, out);
}